// GNNDecoder_10462540333075
// MI455X (gfx1250) — compile-verified
//
#include <hip/hip_runtime.h>

typedef float v2f __attribute__((ext_vector_type(2)));
typedef float v4f __attribute__((ext_vector_type(4)));
typedef float v8f __attribute__((ext_vector_type(8)));

#define NN    100000
#define KK    32
#define LAT   64
#define HID   256
#define OUTD  32
#define NT32  ((NN + 31) / 32)     // 3125 tiles of 32 nodes

#define NL_STRIDE 68               // padded node_lat row stride (banks 4m+32hf)
#define HT_STRIDE 18               // padded h-tile row stride (conflict-free b64)
#define SCR_FLOATS 1152            // per-wave scratch (>= 16*68 = 1088)

// LDS layout (floats):
//   [0,      16384) : W1 B-fragments, frag f=j*16+t, idx = f*64 + lane*2 + p
//   [16384,  24576) : W2 B-fragments, frag f=jo*64+tg, idx = f*64 + lane*2 + p
//   [24576,  33792) : per-wave scratch, 1152 floats each
__launch_bounds__(256, 2)
__global__ void gnn_decoder_kernel(const float* __restrict__ z,
                                   const float* __restrict__ s,
                                   const int*   __restrict__ batch,
                                   const float* __restrict__ W1,
                                   const float* __restrict__ b1,
                                   const float* __restrict__ W2,
                                   const float* __restrict__ b2,
                                   float* __restrict__ out) {
  __shared__ __align__(16) float lds[33792];   // 132 KB
  const int tid  = threadIdx.x;
  const int lane = tid & 31;
  const int wave = tid >> 5;

  // ---- pre-swizzle W1 into WMMA B-fragment order ----
  for (int idx = tid; idx < 16384; idx += 256) {
    const int p   = idx & 1;
    const int ln  = (idx >> 1) & 31;
    const int f   = idx >> 6;
    const int t   = f & 15;
    const int j   = f >> 4;
    const int k   = 4 * t + 2 * (ln >> 4) + p;
    const int col = 16 * j + (ln & 15);
    lds[idx] = W1[k * HID + col];
  }
  // ---- pre-swizzle W2 ----
  for (int idx = tid; idx < 8192; idx += 256) {
    const int p   = idx & 1;
    const int ln  = (idx >> 1) & 31;
    const int f   = idx >> 6;
    const int tg  = f & 63;
    const int jo  = f >> 6;
    const int k   = 4 * tg + 2 * (ln >> 4) + p;
    const int col = 16 * jo + (ln & 15);
    lds[16384 + idx] = W2[k * OUTD + col];
  }
  __syncthreads();

  const int tile = blockIdx.x * 8 + wave;
  if (tile >= NT32) return;
  const int n0 = tile * 32;

  float* scr = &lds[24576 + wave * SCR_FLOATS];

  const int row = lane & 15;
  const int hi  = lane >> 4;

  // ------------- Stage A: node_lat for 2 M-tiles (two 16-node passes) -------
  // A-fragments: af[mt][t] = node_lat rows `row` (+16*mt), cols {4t+2hi,4t+2hi+1}
  v2f af[2][16];
#pragma unroll
  for (int mt = 0; mt < 2; ++mt) {
    const int m    = lane >> 1;          // node within 16-node pass
    const int hf   = lane & 1;           // latent half (32 cols)
    const int node = n0 + mt * 16 + m;
    const int g    = batch[node];
    const float* zbase = z + ((size_t)g * KK) * LAT + hf * 32;

    const v4f* sp = (const v4f*)(s + (size_t)node * KK);
    v4f s4[8];
#pragma unroll
    for (int q = 0; q < 8; ++q) s4[q] = sp[q];

    v4f acc4[8];
#pragma unroll
    for (int q = 0; q < 8; ++q) acc4[q] = (v4f)(0.0f);

#pragma unroll 2
    for (int kq = 0; kq < 8; ++kq) {
      const v4f sq = s4[kq];
#pragma unroll
      for (int ki = 0; ki < 4; ++ki) {
        const float sv = sq[ki];
        const v4f sk = {sv, sv, sv, sv};
        const v4f* zp = (const v4f*)(zbase + (kq * 4 + ki) * LAT);
#pragma unroll
        for (int q = 0; q < 8; ++q)
          acc4[q] = __builtin_elementwise_fma(sk, zp[q], acc4[q]);
      }
    }
    // padded store: bank = 4m + 32hf + ... (at worst 2-way)
    v4f* dst = (v4f*)&scr[m * NL_STRIDE + hf * 32];
#pragma unroll
    for (int q = 0; q < 8; ++q) dst[q] = acc4[q];

    // conflict-free b64 fragment reads (banks 4*row + 4t + 2hi, all distinct)
#pragma unroll
    for (int t = 0; t < 16; ++t)
      af[mt][t] = *(const v2f*)&scr[row * NL_STRIDE + 4 * t + 2 * hi];
  }

  // preload all stage-B biases (no global load inside the j loop)
  float biasv[16];
#pragma unroll
  for (int j = 0; j < 16; ++j) biasv[j] = b1[j * 16 + row];

  // output accumulators seeded with b2
  v8f oc0, oc1, oc2, oc3;
  {
    const float bb0 = b2[row];
    const float bb1 = b2[16 + row];
#pragma unroll
    for (int i = 0; i < 8; ++i) {
      oc0[i] = bb0; oc1[i] = bb1; oc2[i] = bb0; oc3[i] = bb1;
    }
  }

  float* scrA = scr;         // h-tile transpose buffers (16 x HT_STRIDE each)
  float* scrB = scr + 16 * HT_STRIDE;

  // ---------------- Stage B + fused Stage C ----------------
#pragma unroll 1
  for (int j = 0; j < 16; ++j) {
    v8f c0, c1;
#pragma unroll
    for (int i = 0; i < 8; ++i) { c0[i] = biasv[j]; c1[i] = biasv[j]; }

    // each B-fragment load feeds two independent WMMA chains
#pragma unroll
    for (int t = 0; t < 16; ++t) {
      const v2f bf = *(const v2f*)&lds[((j * 16 + t) * 32 + lane) * 2];
      c0 = __builtin_amdgcn_wmma_f32_16x16x4_f32(false, af[0][t], false, bf,
                                                 (short)0, c0, false, false);
      c1 = __builtin_amdgcn_wmma_f32_16x16x4_f32(false, af[1][t], false, bf,
                                                 (short)0, c1, false, false);
    }
#pragma unroll
    for (int i = 0; i < 8; ++i) {
      c0[i] = fmaxf(c0[i], 0.0f);
      c1[i] = fmaxf(c1[i], 0.0f);
    }

    // transpose both 16x16 h-tiles through padded LDS (C-layout -> A-layout)
#pragma unroll
    for (int v = 0; v < 8; ++v) {
      scrA[(v + 8 * hi) * HT_STRIDE + row] = c0[v];
      scrB[(v + 8 * hi) * HT_STRIDE + row] = c1[v];
    }
    v2f ha0[4], ha1[4];
#pragma unroll
    for (int tl = 0; tl < 4; ++tl) {
      ha0[tl] = *(const v2f*)&scrA[row * HT_STRIDE + 4 * tl + 2 * hi];
      ha1[tl] = *(const v2f*)&scrB[row * HT_STRIDE + 4 * tl + 2 * hi];
    }

    // each W2 fragment feeds both M-tiles
#pragma unroll
    for (int tl = 0; tl < 4; ++tl) {
      const int tg = j * 4 + tl;
      const v2f w2a = *(const v2f*)&lds[16384 + (tg * 32 + lane) * 2];
      const v2f w2b = *(const v2f*)&lds[16384 + ((64 + tg) * 32 + lane) * 2];
      oc0 = __builtin_amdgcn_wmma_f32_16x16x4_f32(false, ha0[tl], false, w2a,
                                                  (short)0, oc0, false, false);
      oc1 = __builtin_amdgcn_wmma_f32_16x16x4_f32(false, ha0[tl], false, w2b,
                                                  (short)0, oc1, false, false);
      oc2 = __builtin_amdgcn_wmma_f32_16x16x4_f32(false, ha1[tl], false, w2a,
                                                  (short)0, oc2, false, false);
      oc3 = __builtin_amdgcn_wmma_f32_16x16x4_f32(false, ha1[tl], false, w2b,
                                                  (short)0, oc3, false, false);
    }
  }

  // ---------------- store out[32][32] ----------------
#pragma unroll
  for (int v = 0; v < 8; ++v) {
    const int mrow = v + 8 * hi;
    float* op0 = out + (size_t)(n0 + mrow) * OUTD;
    op0[row]      = oc0[v];
    op0[16 + row] = oc1[v];
    float* op1 = out + (size_t)(n0 + 16 + mrow) * OUTD;
    op1[row]      = oc2[v];
    op1[16 + row] = oc3[v];
  }
}

extern "C" void kernel_launch(void* const* d_in, const int* in_sizes, int n_in,
                              void* d_out, int out_size, void* d_ws, size_t ws_size,
                              hipStream_t stream) {
  const float* z     = (const float*)d_in[0];
  const float* s     = (const float*)d_in[1];
  const int*   batch = (const int*)d_in[2];
  const float* W1    = (const float*)d_in[3];
  const float* b1    = (const float*)d_in[4];
  const float* W2    = (const float*)d_in[5];
  const float* b2    = (const float*)d_in[6];
  float* out = (float*)d_out;

  const int blocks = (NT32 + 7) / 8;   // 8 waves/block, one 32-node tile/wave
  gnn_decoder_kernel<<<blocks, 256, 0, stream>>>(z, s, batch, W1, b1, W2, b2, out);
}